// Model_78056735637556
// MI455X (gfx1250) — compile-verified
//
#include <hip/hip_runtime.h>

typedef __attribute__((ext_vector_type(2))) float v2f;
typedef __attribute__((ext_vector_type(8))) float v8f;

#define NBOX 10
#define NCOORD 4

// Single-wave kernel: the whole problem is 24 floats in / 40 floats out, so it
// is pure latency; one wave32 on one WGP. The outer product boxes = x (.) c is
// computed on the matrix pipe with V_WMMA_F32_16X16X4_F32 (K=0 only), which is
// bit-exact with scalar f32 multiplies under RNE.
__global__ __launch_bounds__(32) void Model_nms_kernel(const float* __restrict__ x,
                                                       const float* __restrict__ c,
                                                       const float* __restrict__ r,
                                                       float* __restrict__ out) {
    __shared__ float sB[16 * 16];   // raw boxes tile from WMMA (16x16, use 10x4)
    __shared__ float sS[16];        // scores
    __shared__ int   sOrder[16];    // argsort(-scores), stable
    __shared__ int   sKeep[16];     // keep flags (sorted-order indexing)
    __shared__ float sBx[16 * 5];   // sorted+offset boxes: x1,y1,x2,y2,area

    const int lane = threadIdx.x;   // wave32: 0..31

    // ---- inputs (predicated selects; EXEC reconverges before WMMA) ----
    const float xv = (lane < NBOX)   ? x[lane] : 0.0f;
    const float cv = (lane < NCOORD) ? c[lane] : 0.0f;
    const float rv = (lane < NBOX)   ? r[lane] : 0.0f;

    // ---- boxes = outer(x, c) via V_WMMA_F32_16X16X4_F32 ----
    // A (16x4, MxK): VGPR0 lanes 0-15 = K=0 column -> A[m][0] = x[m]; rest 0.
    v2f a; a.x = (lane < 16) ? xv : 0.0f; a.y = 0.0f;
    // B (4x16, KxN): VGPR0 lanes 0-15 = K=0 row -> B[0][n] = c[n]; rest 0.
    v2f b; b.x = (lane < 16) ? cv : 0.0f; b.y = 0.0f;
    v8f acc = {};
    // 8 args: (neg_a, A, neg_b, B, c_mod, C, reuse_a, reuse_b)
    v8f d = __builtin_amdgcn_wmma_f32_16x16x4_f32(false, a, false, b,
                                                  (short)0, acc, false, false);

    // D layout: element (M = v + 8*(lane>=16), N = lane&15) lives in VGPR v.
    {
        const int mBase = (lane >> 4) * 8;
        const int col   = lane & 15;
        #pragma unroll
        for (int v = 0; v < 8; ++v)
            sB[(mBase + v) * 16 + col] = d[v];
    }
    if (lane < NBOX) { sS[lane] = rv * xv; sKeep[lane] = 1; }
    __syncthreads();

    // ---- max over the 10x4 boxes (uniform scan; 40 LDS reads) ----
    float mx = sB[0];
    for (int i = 0; i < NBOX; ++i)
        for (int j = 0; j < NCOORD; ++j)
            mx = fmaxf(mx, sB[i * 16 + j]);
    const float offScale = mx + 1.0f;   // fl(max + 1), matches reference

    // ---- stable descending argsort via rank counting ----
    if (lane < NBOX) {
        const float sl = sS[lane];
        int rank = 0;
        for (int j = 0; j < NBOX; ++j) {
            const float sj = sS[j];
            rank += (int)((sj > sl) || ((sj == sl) && (j < lane)));
        }
        sOrder[rank] = lane;            // ranks form a permutation of 0..9
    }
    __syncthreads();

    // ---- sorted + offset boxes (offset uses ORIGINAL row index) ----
    int o = 0;
    if (lane < NBOX) {
        o = sOrder[lane];
        const float off = (float)o * offScale;
        const float x1 = sB[o * 16 + 0] + off;
        const float y1 = sB[o * 16 + 1] + off;
        const float x2 = sB[o * 16 + 2] + off;
        const float y2 = sB[o * 16 + 3] + off;
        sBx[lane * 5 + 0] = x1;
        sBx[lane * 5 + 1] = y1;
        sBx[lane * 5 + 2] = x2;
        sBx[lane * 5 + 3] = y2;
        sBx[lane * 5 + 4] = (x2 - x1) * (y2 - y1);
    }
    __syncthreads();

    float X1 = 0.f, Y1 = 0.f, X2 = 0.f, Y2 = 0.f, AR = 0.f;
    if (lane < NBOX) {
        X1 = sBx[lane * 5 + 0];
        Y1 = sBx[lane * 5 + 1];
        X2 = sBx[lane * 5 + 2];
        Y2 = sBx[lane * 5 + 3];
        AR = sBx[lane * 5 + 4];
    }

    // ---- sequential greedy suppression (thr = 0.0) ----
    // At iteration i, keep[i] is final: only positions > i' get cleared at
    // iteration i', and the end-of-iteration barrier publishes those writes.
    for (int i = 0; i < NBOX; ++i) {
        const int ki = sKeep[i];
        if ((lane < NBOX) && (lane > i) && ki) {
            const float iw = fmaxf(fminf(sBx[i * 5 + 2], X2) - fmaxf(sBx[i * 5 + 0], X1), 0.0f);
            const float ih = fmaxf(fminf(sBx[i * 5 + 3], Y2) - fmaxf(sBx[i * 5 + 1], Y1), 0.0f);
            const float inter = iw * ih;
            const float iou = inter / (sBx[i * 5 + 4] + AR - inter);
            if (iou > 0.0f) sKeep[lane] = 0;   // NaN compares false, as in JAX
        }
        __syncthreads();
    }

    // ---- output: keep[k] ? boxes[order[k]] : 0  (10x4 f32) ----
    if (lane < NBOX) {
        const bool k = sKeep[lane] != 0;
        #pragma unroll
        for (int j = 0; j < NCOORD; ++j)
            out[lane * NCOORD + j] = k ? sB[o * 16 + j] : 0.0f;
    }
}

extern "C" void kernel_launch(void* const* d_in, const int* in_sizes, int n_in,
                              void* d_out, int out_size, void* d_ws, size_t ws_size,
                              hipStream_t stream) {
    const float* x = (const float*)d_in[0];  // (10,)
    const float* c = (const float*)d_in[1];  // (4,)
    const float* r = (const float*)d_in[2];  // (10,)
    float* out = (float*)d_out;              // (10,4) flat = 40 floats
    Model_nms_kernel<<<1, 32, 0, stream>>>(x, c, r, out);
}